// HashEmbedder_2D_34557306864213
// MI455X (gfx1250) — compile-verified
//
#include <hip/hip_runtime.h>

// Instant-NGP 2D hash-grid encoding, forward pass — MI455X (gfx1250).
//
// B = 1e6 points, L = 16 levels, T = 2^19 entries/level, F = 2 features.
//
// Roofline: ~0.64 GFLOP vs ~200 MB mandatory HBM traffic (8 MB x in,
// 64 MB table first-touch, 128 MB out) => pure memory-system problem,
// ~8.6 us floor at 23.3 TB/s. WMMA contributes nothing here; the levers are:
//   * wave32 mapping: thread = (point, level), level = tid & 15, so 16 lanes
//     of a point emit 16 consecutive float2 -> coalesced 8B stores, and the
//     16 lanes' x-loads collapse to one cacheline request.
//   * 64 MB table is L2-resident (192 MB L2): gathers use default RT policy.
//   * 128 MB output stream: non-temporal stores (th:TH_STORE_NT) so the
//     write stream does not evict table lines from WGP$/L2.
//   * 8 MB x stream: non-temporal loads (one-shot data, read once per wave).
//   * 4 independent b64 gathers in flight per thread (MLP) + huge wave count.

#define HG_LEVELS   16
#define HG_LOG2_T   19
#define HG_TBL_MASK ((1u << HG_LOG2_T) - 1u)
#define HG_PRIME1   2654435761u

typedef __attribute__((ext_vector_type(2))) float v2f;

__global__ __launch_bounds__(256) void hashgrid2d_fwd_kernel(
    const float* __restrict__ x,       // [B,2]
    const float* __restrict__ tables,  // [L, T, 2]
    float*       __restrict__ out,     // [B, L*2]
    int B)
{
    const int tid   = blockIdx.x * blockDim.x + threadIdx.x;
    const int point = tid >> 4;   // 16 levels per point
    const int level = tid & 15;
    if (point >= B) return;

    // One-shot stream: NT load keeps x out of the caches the gathers need.
    const v2f p = __builtin_nontemporal_load(
        reinterpret_cast<const v2f*>(x) + point);

    // res = float(int(256 * 2^level)); exact in f32 up to 2^23.
    const float res = (float)(256 << level);

    const float xs = p.x * res;
    const float ys = p.y * res;
    const float fx = floorf(xs);
    const float fy = floorf(ys);
    const float wx = xs - fx;     // compiler fuses to v_fma_f32 (p.x*res - fx)
    const float wy = ys - fy;

    // x in [0,1) => xs in [0,res) => floor >= 0, safe int->uint cast.
    const unsigned ux = (unsigned)(int)fx;
    const unsigned uy = (unsigned)(int)fy;

    // h = cx * PRIME0(=1) ^ cy * PRIME1 ; idx = h & (T-1)
    const unsigned hy0 = uy * HG_PRIME1;
    const unsigned hy1 = (uy + 1u) * HG_PRIME1;
    const unsigned i00 = ( ux        ^ hy0) & HG_TBL_MASK;  // (x0,y0)
    const unsigned i01 = ( ux        ^ hy1) & HG_TBL_MASK;  // (x0,y1)
    const unsigned i10 = ((ux + 1u)  ^ hy0) & HG_TBL_MASK;  // (x1,y0)
    const unsigned i11 = ((ux + 1u)  ^ hy1) & HG_TBL_MASK;  // (x1,y1)

    // 4 independent 8-byte gathers; RT policy -> table stays L2-resident.
    const v2f* __restrict__ tbl =
        reinterpret_cast<const v2f*>(tables) + ((size_t)level << HG_LOG2_T);
    const v2f e00 = tbl[i00];
    const v2f e01 = tbl[i01];
    const v2f e10 = tbl[i10];
    const v2f e11 = tbl[i11];

    // Bilinear: c0 = e00*(1-wx) + e10*wx ; c1 = e01*(1-wx) + e11*wx
    //           o  = c0*(1-wy) + c1*wy
    const float omx = 1.0f - wx;
    const float omy = 1.0f - wy;
    v2f c0, c1, o;
    c0.x = e00.x * omx + e10.x * wx;
    c0.y = e00.y * omx + e10.y * wx;
    c1.x = e01.x * omx + e11.x * wx;
    c1.y = e01.y * omx + e11.y * wx;
    o.x  = c0.x * omy + c1.x * wy;
    o.y  = c0.y * omy + c1.y * wy;

    // out[point, level*2 .. level*2+1]; consecutive lanes -> consecutive 8B.
    // NT store: 128 MB write stream must not evict the table from L2.
    v2f* dst = reinterpret_cast<v2f*>(out) + (size_t)point * HG_LEVELS + level;
    __builtin_nontemporal_store(o, dst);
}

extern "C" void kernel_launch(void* const* d_in, const int* in_sizes, int n_in,
                              void* d_out, int out_size, void* d_ws, size_t ws_size,
                              hipStream_t stream) {
    const float* x      = (const float*)d_in[0];  // [B,2] f32
    const float* tables = (const float*)d_in[1];  // [16, 2^19, 2] f32
    float*       out    = (float*)d_out;          // [B,32] f32

    const int B      = in_sizes[0] / 2;
    const int total  = B * HG_LEVELS;
    const int block  = 256;                       // 8 wave32 waves
    const int blocks = (total + block - 1) / block;

    hashgrid2d_fwd_kernel<<<blocks, block, 0, stream>>>(x, tables, out, B);
}